// Kenn_29661044146691
// MI455X (gfx1250) — compile-verified
//
#include <hip/hip_runtime.h>
#include <hip/hip_bf16.h>
#include <math.h>

#define NN      100000
#define EE      3200000
#define DIN     12
#define HIDDEN  1024
#define OUTC    8
#define ZC      10      // C in reference
#define LITN    22      // 2*C + R
#define NLAYERS 3

typedef float v2f __attribute__((ext_vector_type(2)));
typedef float v8f __attribute__((ext_vector_type(8)));

// ---------------------------------------------------------------------------
// Fused MLP: z[:, 0:8] = relu(X@W1+b1)@W2 + b2 ; z[:,8]=ymin_diff ; z[:,9]=inter
// One wave per 16-row tile, fp32 WMMA 16x16x4 for both GEMMs.
// ---------------------------------------------------------------------------
__global__ __launch_bounds__(256) void mlp_kenn_head(
    const float* __restrict__ X, const float* __restrict__ W1,
    const float* __restrict__ b1, const float* __restrict__ W2,
    const float* __restrict__ b2v, float* __restrict__ z)
{
    __shared__ float lds_h[8][16 * 16];   // per-wave 16x16 h staging (8 KB/block)

    const int lane = threadIdx.x & 31;
    const int wave = threadIdx.x >> 5;
    const int tile = blockIdx.x * 8 + wave;       // wave-uniform
    if (tile >= NN / 16) return;                  // whole wave exits together
    const int r0   = tile * 16;
    const int half = lane >> 4;                   // 0 = lanes 0-15, 1 = lanes 16-31
    const int lidx = lane & 15;

    // A tiles for GEMM1 (X rows r0..r0+15, K=0..11, 3 k-steps of 4)
    // A layout 16x4 f32: lanes0-15 row=lidx K={4s,4s+1}; lanes16-31 K={4s+2,4s+3}
    v2f a1[3];
    {
        const float* xr = X + (size_t)(r0 + lidx) * DIN;
        #pragma unroll
        for (int s = 0; s < 3; ++s) {
            const int k = 4 * s + half * 2;
            a1[s].x = xr[k + 0];
            a1[s].y = xr[k + 1];
        }
    }

    v8f   c2 = {};                                // 16x16 z accumulator (cols 0-7 valid)
    float* hl = &lds_h[wave][0];

    for (int n = 0; n < HIDDEN / 16; ++n) {
        // ---- GEMM1: h(16x16) = A(16x12) x W1[0:12, 16n:16n+16]
        v8f c1 = {};
        #pragma unroll
        for (int s = 0; s < 3; ++s) {
            const int kb = 4 * s + half * 2;      // B 4x16: K split across lane halves
            v2f b;
            b.x = W1[(size_t)(kb + 0) * HIDDEN + n * 16 + lidx];
            b.y = W1[(size_t)(kb + 1) * HIDDEN + n * 16 + lidx];
            c1 = __builtin_amdgcn_wmma_f32_16x16x4_f32(false, a1[s], false, b,
                                                       (short)0, c1, false, false);
        }
        // bias + relu, stage to LDS in row-major (C layout: vgpr v -> M=v+8*half, N=lidx)
        const float bias = b1[n * 16 + lidx];
        #pragma unroll
        for (int v = 0; v < 8; ++v) {
            float h = c1[v] + bias;
            h = h > 0.f ? h : 0.f;
            hl[(v + half * 8) * 16 + lidx] = h;
        }
        // ---- GEMM2 partial: c2 += h(16x16) x W2[16n:16n+16, 0:8 (pad 16)]
        // LDS ops are in-order within a wave: loads below see the stores above.
        #pragma unroll
        for (int s = 0; s < 4; ++s) {
            const int ka = 4 * s + half * 2;
            v2f a2;
            a2.x = hl[lidx * 16 + ka + 0];        // consecutive -> ds_load_b64
            a2.y = hl[lidx * 16 + ka + 1];
            const int kb  = 16 * n + 4 * s + half * 2;
            const int col = lidx & 7;             // clamped load + select, no EXEC change
            const float w0 = W2[(size_t)(kb + 0) * OUTC + col];
            const float w1 = W2[(size_t)(kb + 1) * OUTC + col];
            v2f b;
            b.x = (lidx < OUTC) ? w0 : 0.f;
            b.y = (lidx < OUTC) ? w1 : 0.f;
            c2 = __builtin_amdgcn_wmma_f32_16x16x4_f32(false, a2, false, b,
                                                       (short)0, c2, false, false);
        }
    }

    // ---- write z rows: cols 0-7 = c2 + b2 ; col 8 = ymin_diff ; col 9 = intersection
    #pragma unroll
    for (int v = 0; v < 8; ++v) {
        const int row = r0 + v + half * 8;
        if (lidx < OUTC) {
            z[(size_t)row * ZC + lidx] = c2[v] + b2v[lidx];
        } else if (lidx == 8) {
            const float* f = X + (size_t)row * DIN;
            z[(size_t)row * ZC + 8] = (f[2] - f[10]) * 10.f;
        } else if (lidx == 9) {
            const float* f = X + (size_t)row * DIN;
            const bool m = (f[0] <= f[5]) && (f[1] >= f[4]) &&
                           (f[2] <= f[11]) && (f[3] >= f[10]);
            z[(size_t)row * ZC + 9] = m ? 5.f : -5.f;
        }
    }
}

// ---------------------------------------------------------------------------
// One KENN layer: per-edge 22-way signed softmax, scatter-add into zout.
// zout must be pre-initialized to a copy of zin (gather-then-scatter semantics).
// ---------------------------------------------------------------------------
__global__ __launch_bounds__(256) void kenn_edge(
    const float* __restrict__ zin, float* __restrict__ zout,
    const float* __restrict__ rel,
    const long long* __restrict__ sx, const long long* __restrict__ sy,
    const float* __restrict__ clause_w, int layer)
{
    const int e = blockIdx.x * blockDim.x + threadIdx.x;
    if (e >= EE) return;

    const float w  = clause_w[layer];
    const float sp = (w > 20.f) ? w : log1pf(__expf(w));   // softplus

    const long long a = sx[e];
    const long long b = sy[e];
    const float* za = zin + (size_t)a * ZC;
    const float* zb = zin + (size_t)b * ZC;

    float g[LITN];
    #pragma unroll
    for (int i = 0; i < ZC; ++i) g[i] = za[i];
    #pragma unroll
    for (int i = 0; i < ZC; ++i) g[ZC + i] = zb[i];
    g[20] = rel[(size_t)e * 2 + 0];
    g[21] = rel[(size_t)e * 2 + 1];

    float m = -3.4e38f;
    float sg[LITN];
    #pragma unroll
    for (int i = 0; i < LITN; ++i) {
        const float s = (i & 1) ? -1.f : 1.f;
        sg[i] = s * g[i];
        m = fmaxf(m, sg[i]);
    }
    float sum = 0.f;
    float ex[LITN];
    #pragma unroll
    for (int i = 0; i < LITN; ++i) {
        ex[i] = __expf(sg[i] - m);
        sum += ex[i];
    }
    const float scale = sp / sum;

    float* oa = zout + (size_t)a * ZC;
    float* ob = zout + (size_t)b * ZC;
    #pragma unroll
    for (int i = 0; i < ZC; ++i) {
        const float s = (i & 1) ? -1.f : 1.f;
        atomicAdd(&oa[i], scale * ex[i] * s);
    }
    #pragma unroll
    for (int i = 0; i < ZC; ++i) {
        const float s = (i & 1) ? -1.f : 1.f;
        atomicAdd(&ob[i], scale * ex[ZC + i] * s);
    }
}

// ---------------------------------------------------------------------------
// Output: za, softmax(za), zb, softmax(zb) concatenated flat in return order.
// ---------------------------------------------------------------------------
__global__ __launch_bounds__(256) void kenn_output(
    const float* __restrict__ z, float* __restrict__ out)
{
    const int i = blockIdx.x * blockDim.x + threadIdx.x;
    if (i >= NN) return;
    const float* zr = z + (size_t)i * ZC;

    float za[4], zb[4];
    #pragma unroll
    for (int j = 0; j < 4; ++j) { za[j] = zr[j]; zb[j] = zr[4 + j]; }

    float ma = fmaxf(fmaxf(za[0], za[1]), fmaxf(za[2], za[3]));
    float mb = fmaxf(fmaxf(zb[0], zb[1]), fmaxf(zb[2], zb[3]));
    float ea[4], eb[4], sa = 0.f, sb = 0.f;
    #pragma unroll
    for (int j = 0; j < 4; ++j) { ea[j] = __expf(za[j] - ma); sa += ea[j]; }
    #pragma unroll
    for (int j = 0; j < 4; ++j) { eb[j] = __expf(zb[j] - mb); sb += eb[j]; }
    const float ra = 1.f / sa, rb = 1.f / sb;

    float* o0 = out;
    float* o1 = out + (size_t)4 * NN;
    float* o2 = out + (size_t)8 * NN;
    float* o3 = out + (size_t)12 * NN;
    #pragma unroll
    for (int j = 0; j < 4; ++j) {
        o0[(size_t)i * 4 + j] = za[j];
        o1[(size_t)i * 4 + j] = ea[j] * ra;
        o2[(size_t)i * 4 + j] = zb[j];
        o3[(size_t)i * 4 + j] = eb[j] * rb;
    }
}

extern "C" void kernel_launch(void* const* d_in, const int* in_sizes, int n_in,
                              void* d_out, int out_size, void* d_ws, size_t ws_size,
                              hipStream_t stream) {
    const float*     features  = (const float*)d_in[0];
    const float*     relations = (const float*)d_in[1];
    const float*     W1        = (const float*)d_in[2];
    const float*     b1        = (const float*)d_in[3];
    const float*     W2        = (const float*)d_in[4];
    const float*     b2        = (const float*)d_in[5];
    const float*     clause_w  = (const float*)d_in[6];
    const long long* sx        = (const long long*)d_in[7];
    const long long* sy        = (const long long*)d_in[8];
    float*           out       = (float*)d_out;

    float* z0 = (float*)d_ws;                    // N*10 f32 = 4 MB
    float* z1 = z0 + (size_t)NN * ZC;            // second 4 MB

    // MLP head (WMMA): 6250 tiles, 8 waves/block
    mlp_kenn_head<<<(NN / 16 + 7) / 8, 256, 0, stream>>>(features, W1, b1, W2, b2, z0);

    float* zin = z0;
    float* zout = z1;
    for (int l = 0; l < NLAYERS; ++l) {
        hipMemcpyAsync(zout, zin, (size_t)NN * ZC * sizeof(float),
                       hipMemcpyDeviceToDevice, stream);
        kenn_edge<<<(EE + 255) / 256, 256, 0, stream>>>(zin, zout, relations,
                                                        sx, sy, clause_w, l);
        float* t = zin; zin = zout; zout = t;
    }

    kenn_output<<<(NN + 255) / 256, 256, 0, stream>>>(zin, out);
}